// GCN_15710990369132
// MI455X (gfx1250) — compile-verified
//
#include <hip/hip_runtime.h>
#include <math.h>

typedef float v2f __attribute__((ext_vector_type(2)));
typedef float v8f __attribute__((ext_vector_type(8)));

#define NNODES 100000
#define NEDGES 1000000
#define HDIM   64
#define NCLASS 40

// ---------------- degree / dinv ----------------
__global__ __launch_bounds__(256) void k_init_deg(float* deg, int n) {
    int i = blockIdx.x * blockDim.x + threadIdx.x;
    if (i < n) deg[i] = 1.0f;  // self-loop
}

__global__ __launch_bounds__(256) void k_count_deg(const long long* __restrict__ ei,
                                                   float* __restrict__ deg, int e) {
    int i = blockIdx.x * blockDim.x + threadIdx.x;
    if (i < e) atomicAdd(&deg[(size_t)ei[(size_t)e + i]], 1.0f);
}

__global__ __launch_bounds__(256) void k_rsqrt(float* deg, int n) {
    int i = blockIdx.x * blockDim.x + threadIdx.x;
    if (i < n) deg[i] = rsqrtf(deg[i]);
}

__global__ __launch_bounds__(256) void k_zero(float* p, long long n) {
    long long i = (long long)blockIdx.x * blockDim.x + threadIdx.x;
    if (i < n) p[i] = 0.0f;
}

// ---------------- fp32 WMMA GEMM: out[M,64] = X[M,64] @ W[64,64] (+ bias) ----------------
// One wave computes a 16-row x 64-col strip using V_WMMA_F32_16X16X4_F32.
__global__ __launch_bounds__(256) void k_gemm64(const float* __restrict__ X,
                                                const float* __restrict__ W,
                                                const float* __restrict__ bias,
                                                float* __restrict__ out,
                                                int nrows, int use_bias) {
    const int lane  = threadIdx.x & 31;
    const int wave  = threadIdx.x >> 5;                  // 0..7
    const int m0    = (blockIdx.x * 8 + wave) * 16;      // wave-uniform
    if (m0 >= nrows) return;                             // whole wave exits together
    const int mrow  = lane & 15;                         // row within tile / col within tile
    const int khalf = (lane >> 4) << 1;                  // 0 or 2
    const int rhalf = (lane >> 4) << 3;                  // 0 or 8 (C-matrix row offset)

    v8f c0 = {}, c1 = {}, c2 = {}, c3 = {};
    const float* xrow = X + (size_t)(m0 + mrow) * HDIM;

#pragma unroll
    for (int kk = 0; kk < 16; ++kk) {
        const int k = kk * 4 + khalf;
        v2f a = *(const v2f*)(xrow + k);                 // A: lane holds X[m, k], X[m, k+1]
        const float* wr0 = W + (size_t)k * HDIM;
        const float* wr1 = wr0 + HDIM;
#pragma unroll
        for (int nt = 0; nt < 4; ++nt) {
            const int ncol = nt * 16 + mrow;
            v2f b;
            b.x = wr0[ncol];                             // B: W[k,   n]
            b.y = wr1[ncol];                             // B: W[k+1, n]
            v8f acc = (nt == 0) ? c0 : (nt == 1) ? c1 : (nt == 2) ? c2 : c3;
            acc = __builtin_amdgcn_wmma_f32_16x16x4_f32(false, a, false, b,
                                                        (short)0, acc, false, false);
            if (nt == 0) c0 = acc; else if (nt == 1) c1 = acc;
            else if (nt == 2) c2 = acc; else c3 = acc;
        }
    }

#pragma unroll
    for (int nt = 0; nt < 4; ++nt) {
        const int ncol = nt * 16 + mrow;
        v8f acc = (nt == 0) ? c0 : (nt == 1) ? c1 : (nt == 2) ? c2 : c3;
        float bv = use_bias ? bias[ncol] : 0.0f;
#pragma unroll
        for (int r = 0; r < 8; ++r) {
            const int m = m0 + r + rhalf;
            out[(size_t)m * HDIM + ncol] = acc[r] + bv;
        }
    }
}

// ---------------- edge scatter: agg[dst,:] += h[src,:] * dinv[src]*dinv[dst] ----------------
// Thread t -> (edge e = t/64, feature j = t%64): per-wave contiguous 128B load + 128B atomics.
__global__ __launch_bounds__(256) void k_scatter(const long long* __restrict__ ei,
                                                 const float* __restrict__ h,
                                                 const float* __restrict__ dinv,
                                                 float* __restrict__ agg, int e) {
    long long t = (long long)blockIdx.x * blockDim.x + threadIdx.x;
    if (t >= (long long)e * HDIM) return;
    const int  j = (int)(t & (HDIM - 1));
    const long long idx = t >> 6;
    const long long s = ei[idx];
    const long long d = ei[(size_t)e + idx];
    const float norm = dinv[s] * dinv[d];
    atomicAdd(&agg[(size_t)d * HDIM + j], h[(size_t)s * HDIM + j] * norm);
}

// ---------------- combine: v = agg + h*dinv^2 + b ; emit pre-act and/or tanh ----------------
__global__ __launch_bounds__(256) void k_combine(const float* __restrict__ agg,
                                                 const float* __restrict__ h,
                                                 const float* __restrict__ dinv,
                                                 const float* __restrict__ bias,
                                                 float* __restrict__ out_pre,
                                                 float* __restrict__ out_act,
                                                 int write_pre, long long nelem) {
    long long t = (long long)blockIdx.x * blockDim.x + threadIdx.x;
    if (t >= nelem) return;
    const long long n = t >> 6;
    const int j = (int)(t & (HDIM - 1));
    const float di = dinv[n];
    const float v = agg[t] + h[t] * di * di + bias[j];
    if (write_pre) out_pre[t] = v;
    out_act[t] = tanhf(v);
}

// ---------------- pad Wp2 (64x40 -> 64x64) and bp2 (40 -> 64) with zeros ----------------
__global__ __launch_bounds__(256) void k_pad_wp2(const float* __restrict__ Wp2,
                                                 const float* __restrict__ bp2,
                                                 float* __restrict__ Wpad,
                                                 float* __restrict__ bpad) {
    int t = blockIdx.x * blockDim.x + threadIdx.x;
    if (t < HDIM * HDIM) {
        int k = t >> 6, c = t & 63;
        Wpad[t] = (c < NCLASS) ? Wp2[k * NCLASS + c] : 0.0f;
    }
    if (t < HDIM) bpad[t] = (t < NCLASS) ? bp2[t] : 0.0f;
}

// ---------------- log_softmax over 40 valid cols of a [N,64] buffer, wave per row ----------
__global__ __launch_bounds__(256) void k_logsoftmax(const float* __restrict__ logits,
                                                    float* __restrict__ out, int n) {
    const int lane = threadIdx.x & 31;
    const int wave = threadIdx.x >> 5;
    const int row  = blockIdx.x * 8 + wave;
    if (row >= n) return;
    const float* rp = logits + (size_t)row * HDIM;
    const float v0 = (lane < NCLASS) ? rp[lane] : -INFINITY;
    const float v1 = (lane + 32 < NCLASS) ? rp[lane + 32] : -INFINITY;
    float m = fmaxf(v0, v1);
#pragma unroll
    for (int off = 16; off > 0; off >>= 1) m = fmaxf(m, __shfl_xor(m, off, 32));
    float s = ((lane < NCLASS) ? expf(v0 - m) : 0.0f) +
              ((lane + 32 < NCLASS) ? expf(v1 - m) : 0.0f);
#pragma unroll
    for (int off = 16; off > 0; off >>= 1) s += __shfl_xor(s, off, 32);
    const float lse = m + logf(s);
    if (lane < NCLASS)      out[(size_t)row * NCLASS + lane]      = v0 - lse;
    if (lane + 32 < NCLASS) out[(size_t)row * NCLASS + lane + 32] = v1 - lse;
}

extern "C" void kernel_launch(void* const* d_in, const int* in_sizes, int n_in,
                              void* d_out, int out_size, void* d_ws, size_t ws_size,
                              hipStream_t stream) {
    (void)in_sizes; (void)n_in; (void)out_size; (void)ws_size;
    const float*     x   = (const float*)d_in[0];
    const long long* ei  = (const long long*)d_in[1];
    const float*     W0  = (const float*)d_in[2];
    const float*     b0  = (const float*)d_in[3];
    const float*     W1  = (const float*)d_in[4];
    const float*     b1  = (const float*)d_in[5];
    const float*     W2  = (const float*)d_in[6];
    const float*     b2  = (const float*)d_in[7];
    const float*     Wp1 = (const float*)d_in[8];
    const float*     bp1 = (const float*)d_in[9];
    const float*     Wp2 = (const float*)d_in[10];
    const float*     bp2 = (const float*)d_in[11];

    float* emb = (float*)d_out;                       // [N, 64]
    float* lsm = emb + (size_t)NNODES * HDIM;         // [N, 40]

    float* ws     = (float*)d_ws;
    float* dinv   = ws;                               // N (deg -> dinv in place)
    float* bufT   = ws + 100352;                      // N*64 (h = X@W, also MLP temps)
    float* bufH   = bufT + (size_t)NNODES * HDIM;     // N*64 (activations)
    float* bufAgg = bufH + (size_t)NNODES * HDIM;     // N*64 (scatter accum / logits)
    float* Wpad   = bufAgg + (size_t)NNODES * HDIM;   // 64*64
    float* bpad   = Wpad + HDIM * HDIM;               // 64

    const long long NE = (long long)NNODES * HDIM;    // 6,400,000
    const int gN   = (NNODES + 255) / 256;
    const int gE   = (NEDGES + 255) / 256;
    const int gNE  = (int)((NE + 255) / 256);
    const int gEH  = (int)(((long long)NEDGES * HDIM + 255) / 256);
    const int gMM  = (NNODES + 127) / 128;            // 128 rows per block (8 waves x 16)
    const int gSM  = (NNODES + 7) / 8;                // 8 rows per block (wave per row)

    // degrees -> dinv
    k_init_deg<<<gN, 256, 0, stream>>>(dinv, NNODES);
    k_count_deg<<<gE, 256, 0, stream>>>(ei, dinv, NEDGES);
    k_rsqrt<<<gN, 256, 0, stream>>>(dinv, NNODES);

    const float* Ws[3] = { W0, W1, W2 };
    const float* bs[3] = { b0, b1, b2 };
    const float* Xin = x;
    for (int l = 0; l < 3; ++l) {
        k_gemm64<<<gMM, 256, 0, stream>>>(Xin, Ws[l], nullptr, bufT, NNODES, 0);
        k_zero<<<gNE, 256, 0, stream>>>(bufAgg, NE);
        k_scatter<<<gEH, 256, 0, stream>>>(ei, bufT, dinv, bufAgg, NEDGES);
        k_combine<<<gNE, 256, 0, stream>>>(bufAgg, bufT, dinv, bs[l],
                                           (l == 2) ? emb : nullptr, bufH,
                                           (l == 2) ? 1 : 0, NE);
        Xin = bufH;
    }

    // post_mp MLP: (tanh(emb) @ Wp1 + bp1) @ Wp2pad + bp2pad, then log_softmax over 40
    k_gemm64<<<gMM, 256, 0, stream>>>(bufH, Wp1, bp1, bufT, NNODES, 1);
    k_pad_wp2<<<16, 256, 0, stream>>>(Wp2, bp2, Wpad, bpad);
    k_gemm64<<<gMM, 256, 0, stream>>>(bufT, Wpad, bpad, bufAgg, NNODES, 1);
    k_logsoftmax<<<gSM, 256, 0, stream>>>(bufAgg, lsm, NNODES);
}